// CLIPVisionTower_VisionZip_TextAware_74062416053278
// MI455X (gfx1250) — compile-verified
//
#include <hip/hip_runtime.h>
#include <math.h>

// Problem constants (fixed by the reference setup)
#define BB     32
#define HH     16
#define LL     577
#define DD     1024
#define CKK    64
#define LP     576     // patches
#define DOMK   54      // top-k dominant patches
#define NDOM   55      // + CLS
#define NR     522     // remaining tokens
#define CTX    10      // contextual targets
#define STEPC  52      // Nr // contextual
#define NMERGE 512     // tokens merged into targets
#define LOUT   65      // output rows per batch

typedef __attribute__((ext_vector_type(2))) float v2f;
typedef __attribute__((ext_vector_type(8))) float v8f;

// Map merge index m (0..511) -> rank r in the remaining-token list (0..521),
// skipping target ranks {0, 52, ..., 468}.
__device__ __forceinline__ int remain_rank(int m) {
    return (m < 459) ? (m + m / 51 + 1) : (m + 10);
}

// ---------------------------------------------------------------------------
// Kernel 1: per-batch score = 0.5*z(Sd) + 0.5*z(cos); top-54 selection;
// emit sorted dominant index list (incl. CLS=0) and remaining index list.
// One block per batch, 576 threads (one per patch).
// ---------------------------------------------------------------------------
__global__ __launch_bounds__(576) void score_select_kernel(
    const float* __restrict__ attn, const float* __restrict__ metric,
    const float* __restrict__ text, int* __restrict__ dom, int* __restrict__ rem)
{
    __shared__ float sc[LP];
    __shared__ float red[1024];
    __shared__ int   ridx[1024];
    __shared__ int   sel[LP];

    const int b = blockIdx.x;
    const int t = threadIdx.x;   // 0..575, == patch index j

    // ||text_emb[b]|| (redundant per thread; 64 cached floats, negligible)
    float tn = 0.f;
    for (int c = 0; c < CKK; ++c) { float v = text[b * CKK + c]; tn += v * v; }
    tn = sqrtf(tn) + 1e-12f;

    // Sd[j] = sum_h attn[b,h,0,1+j]
    float sd = 0.f;
    {
        const float* ap = attn + (size_t)b * HH * LL * LL + 1 + t;
        for (int h = 0; h < HH; ++h) sd += ap[(size_t)h * LL * LL];
    }
    // cos[j] = <metric[b,1+j,:], text[b,:]> / ((||m||+eps)(||t||+eps))
    float cs;
    {
        const float* mp = metric + ((size_t)b * LL + 1 + t) * CKK;
        float dot = 0.f, nn = 0.f;
        for (int c = 0; c < CKK; ++c) {
            float v = mp[c];
            dot += v * text[b * CKK + c];
            nn  += v * v;
        }
        cs = dot / ((sqrtf(nn) + 1e-12f) * tn);
    }

    // --- block sum helper (inlined twice per z-score) ---
    auto block_sum = [&](float v) -> float {
        red[t] = v;
        if (t < 1024 - LP) red[LP + t] = 0.f;
        __syncthreads();
        for (int s = 512; s > 0; s >>= 1) {
            if (t < s) red[t] += red[t + s];
            __syncthreads();
        }
        float r = red[0];
        __syncthreads();
        return r;
    };

    // z-score of Sd (ddof=1)
    float m1 = block_sum(sd) / (float)LP;
    float v1 = block_sum((sd - m1) * (sd - m1)) / (float)(LP - 1);
    float z1 = (sd - m1) / (sqrtf(v1) + 1e-6f);
    // z-score of cos
    float m2 = block_sum(cs) / (float)LP;
    float v2 = block_sum((cs - m2) * (cs - m2)) / (float)(LP - 1);
    float z2 = (cs - m2) / (sqrtf(v2) + 1e-6f);

    sc[t]  = 0.5f * z1 + 0.5f * z2;
    sel[t] = 0;
    __syncthreads();

    // Iterative top-54: argmax with ties -> lowest index (matches lax.top_k set)
    for (int it = 0; it < DOMK; ++it) {
        red[t]  = sel[t] ? -INFINITY : sc[t];
        ridx[t] = t;
        if (t < 1024 - LP) { red[LP + t] = -INFINITY; ridx[LP + t] = 1 << 30; }
        __syncthreads();
        for (int s = 512; s > 0; s >>= 1) {
            if (t < s) {
                float a = red[t], c = red[t + s];
                if (c > a || (c == a && ridx[t + s] < ridx[t])) {
                    red[t] = c; ridx[t] = ridx[t + s];
                }
            }
            __syncthreads();
        }
        if (t == 0) sel[ridx[0]] = 1;
        __syncthreads();
    }

    // Emit sorted index lists (cheap serial scan; 2 x 576 iters)
    if (t == 0) {
        int nd = 0;
        dom[b * NDOM + nd++] = 0;                      // CLS
        for (int j = 0; j < LP; ++j)
            if (sel[j]) dom[b * NDOM + nd++] = j + 1;
        int nr = 0;
        for (int j = 0; j < LP; ++j)
            if (!sel[j]) rem[b * NR + nr++] = j + 1;
    }
}

// ---------------------------------------------------------------------------
// Kernel 2: gather dominant tokens -> out[:, 0:55, :]
// ---------------------------------------------------------------------------
__global__ __launch_bounds__(256) void gather_dom_kernel(
    const float* __restrict__ hs, const int* __restrict__ dom,
    float* __restrict__ out)
{
    const int i = blockIdx.x;   // 0..54
    const int b = blockIdx.y;   // 0..31
    const int tok = dom[b * NDOM + i];
    const float4* src = (const float4*)(hs + ((size_t)b * LL + tok) * DD);
    float4*       dst = (float4*)(out + ((size_t)b * LOUT + i) * DD);
    for (int k = threadIdx.x; k < DD / 4; k += blockDim.x) dst[k] = src[k];
}

// ---------------------------------------------------------------------------
// Kernel 3: sim = merge(512x64) x targets^T(64x10) via V_WMMA_F32_16X16X4_F32,
// argmax over 10 targets -> assign[512]. One block per batch, 512 threads
// (16 waves; each wave owns 2 M-tiles of 16 rows, N padded 10->16).
// ---------------------------------------------------------------------------
__global__ __launch_bounds__(512) void sim_assign_kernel(
    const float* __restrict__ metric, const int* __restrict__ rem,
    int* __restrict__ assign)
{
    __shared__ float Tt[CKK][16];           // targets^T, zero-padded columns
    __shared__ float simbuf[16][16][17];    // per-wave C tile (padded rows)

    const int b    = blockIdx.x;
    const int t    = threadIdx.x;
    const int wave = t >> 5;
    const int lane = t & 31;

    // zero-pad target matrix
    for (int i = t; i < CKK * 16; i += 512) ((float*)Tt)[i] = 0.f;
    __syncthreads();

    // Waves 0..9: normalize one target row each into B-matrix layout Tt[c][k]
    if (wave < CTX) {
        const int tok = rem[b * NR + STEPC * wave];
        const float* mp = metric + ((size_t)b * LL + tok) * CKK;
        float v0 = mp[lane], v1 = mp[lane + 32];
        float ss = v0 * v0 + v1 * v1;
        for (int off = 16; off > 0; off >>= 1) ss += __shfl_xor(ss, off, 32);
        float inv = 1.f / (sqrtf(ss) + 1e-12f);
        Tt[lane][wave]      = v0 * inv;
        Tt[lane + 32][wave] = v1 * inv;
    }
    __syncthreads();

    const int mrow = lane & 15;               // A row / B column in half-wave
    const int kb   = (lane < 16) ? 0 : 2;     // K sub-offset for this half

    for (int tile = wave; tile < NMERGE / 16; tile += 16) {   // exactly 2 iters/wave
        // Load + normalize this lane's half of row `mrow` of the tile
        const int m_g = tile * 16 + mrow;
        const int tok = rem[b * NR + remain_rank(m_g)];
        const float* mp = metric + ((size_t)b * LL + tok) * CKK;

        float areg[32];
        float ss = 0.f;
        #pragma unroll
        for (int g = 0; g < 16; ++g) {
            float x = mp[4 * g + kb];
            float y = mp[4 * g + kb + 1];
            areg[2 * g]     = x;
            areg[2 * g + 1] = y;
            ss += x * x + y * y;
        }
        ss += __shfl_xor(ss, 16, 32);         // combine row halves
        float inv = 1.f / (sqrtf(ss) + 1e-12f);
        #pragma unroll
        for (int g = 0; g < 32; ++g) areg[g] *= inv;

        // 16 x (16x16x4 f32 WMMA) over K=64
        v8f c = {};
        #pragma unroll
        for (int step = 0; step < 16; ++step) {
            v2f a, bv;
            a.x  = areg[2 * step];
            a.y  = areg[2 * step + 1];
            bv.x = Tt[4 * step + kb][mrow];
            bv.y = Tt[4 * step + kb + 1][mrow];
            c = __builtin_amdgcn_wmma_f32_16x16x4_f32(
                    false, a, false, bv, (short)0, c, false, false);
        }

        // Spill C tile (row = vgpr + 8*(lane>=16), col = lane%16) and argmax
        const int rbase = (lane < 16) ? 0 : 8;
        #pragma unroll
        for (int v = 0; v < 8; ++v) simbuf[wave][rbase + v][mrow] = c[v];
        __syncthreads();
        if (lane < 16) {
            float best = simbuf[wave][lane][0];
            int   bi   = 0;
            #pragma unroll
            for (int n = 1; n < CTX; ++n) {
                float vv = simbuf[wave][lane][n];
                if (vv > best) { best = vv; bi = n; }   // first max, like argmax
            }
            assign[b * NMERGE + tile * 16 + lane] = bi;
        }
        __syncthreads();
    }
}

// ---------------------------------------------------------------------------
// Kernel 4: scatter-mean merge + write contextual tokens out[:, 55:65, :].
// One block per batch, 1024 threads (one hidden column each).
// ---------------------------------------------------------------------------
__global__ __launch_bounds__(1024) void merge_kernel(
    const float* __restrict__ hs, const int* __restrict__ rem,
    const int* __restrict__ assign, float* __restrict__ out)
{
    __shared__ int cnt[CTX];
    __shared__ int asg[NMERGE];
    __shared__ int toks[NMERGE];

    const int b = blockIdx.x;
    const int d = threadIdx.x;

    if (d < CTX) cnt[d] = 0;
    __syncthreads();
    if (d < NMERGE) {
        int a = assign[b * NMERGE + d];
        asg[d]  = a;
        toks[d] = rem[b * NR + remain_rank(d)];
        atomicAdd(&cnt[a], 1);
    }
    __syncthreads();

    float acc[CTX];
    #pragma unroll
    for (int k = 0; k < CTX; ++k) acc[k] = 0.f;

    for (int m = 0; m < NMERGE; ++m) {
        float hv = hs[((size_t)b * LL + toks[m]) * DD + d];   // coalesced
        int   a  = asg[m];
        #pragma unroll
        for (int k = 0; k < CTX; ++k)
            if (a == k) acc[k] += hv;                         // predicated adds
    }

    #pragma unroll
    for (int k = 0; k < CTX; ++k) {
        int   tok = rem[b * NR + STEPC * k];                  // tgt token
        float cN  = (float)(cnt[k] > 1 ? cnt[k] : 1);
        out[((size_t)b * LOUT + NDOM + k) * DD + d] =
            hs[((size_t)b * LL + tok) * DD + d] + acc[k] / cN;
    }
}

// ---------------------------------------------------------------------------
extern "C" void kernel_launch(void* const* d_in, const int* in_sizes, int n_in,
                              void* d_out, int out_size, void* d_ws, size_t ws_size,
                              hipStream_t stream)
{
    const float* attn   = (const float*)d_in[0];
    const float* hs     = (const float*)d_in[1];
    const float* metric = (const float*)d_in[2];
    const float* text   = (const float*)d_in[3];
    float* out = (float*)d_out;

    int* dom    = (int*)d_ws;            // B*55
    int* rem    = dom + BB * NDOM;       // B*522
    int* assign = rem + BB * NR;         // B*512

    score_select_kernel<<<BB, 576, 0, stream>>>(attn, metric, text, dom, rem);

    dim3 g2(NDOM, BB);
    gather_dom_kernel<<<g2, 256, 0, stream>>>(hs, dom, out);

    sim_assign_kernel<<<BB, 512, 0, stream>>>(metric, rem, assign);

    merge_kernel<<<BB, 1024, 0, stream>>>(hs, rem, assign, out);
}